// GCN_30691836297928
// MI455X (gfx1250) — compile-verified
//
#include <hip/hip_runtime.h>
#include <hip/hip_bf16.h>
#include <stdint.h>

typedef __attribute__((ext_vector_type(2))) float v2f;
typedef __attribute__((ext_vector_type(8))) float v8f;

#define FEATS 128
#define LDA 132   // 128 + 4 pad: conflict-free A-fragment column reads, 16B-aligned rows

// ---------------- degree histogram ----------------
__global__ void k_degrees(const int* __restrict__ src, const int* __restrict__ dst,
                          int* __restrict__ dout, int* __restrict__ din, int E) {
  int i = blockIdx.x * blockDim.x + threadIdx.x;
  if (i < E) {
    atomicAdd(&dout[src[i]], 1);
    atomicAdd(&din[dst[i]], 1);
  }
}

__global__ void k_norms(const int* __restrict__ dout, const int* __restrict__ din,
                        float* __restrict__ nsrc, float* __restrict__ ndst, int n) {
  int i = blockIdx.x * blockDim.x + threadIdx.x;
  if (i < n) {
    nsrc[i] = rsqrtf(fmaxf((float)dout[i], 1.0f));
    ndst[i] = rsqrtf(fmaxf((float)din[i], 1.0f));
  }
}

// ---------------- exclusive scan (3-phase) for CSR row_ptr ----------------
__global__ void k_scan1(const int* __restrict__ counts, int* __restrict__ incl,
                        int* __restrict__ bsum, int n) {
  __shared__ int tmp[256];
  int tid = threadIdx.x;
  int i = blockIdx.x * 256 + tid;
  int v = (i < n) ? counts[i] : 0;
  tmp[tid] = v;
  __syncthreads();
  for (int off = 1; off < 256; off <<= 1) {
    int t = (tid >= off) ? tmp[tid - off] : 0;
    __syncthreads();
    tmp[tid] += t;
    __syncthreads();
  }
  if (i < n) incl[i] = tmp[tid];
  if (tid == 255) bsum[blockIdx.x] = tmp[255];
}

__global__ void k_scan2(int* __restrict__ bsum, int nb) {
  if (threadIdx.x == 0 && blockIdx.x == 0) {
    int run = 0;
    for (int b = 0; b < nb; ++b) {
      int s = bsum[b];
      bsum[b] = run;   // exclusive block offsets
      run += s;
    }
  }
}

__global__ void k_scan3(const int* __restrict__ counts, const int* __restrict__ incl,
                        const int* __restrict__ bsum, int* __restrict__ rowptr,
                        int* __restrict__ cursor, int n) {
  int i = blockIdx.x * 256 + threadIdx.x;
  if (i < n) {
    int tot = incl[i] + bsum[blockIdx.x];   // inclusive global scan
    rowptr[i + 1] = tot;
    cursor[i] = tot - counts[i];            // exclusive start for counting sort
    if (i == 0) rowptr[0] = 0;
  }
}

__global__ void k_scatter(const int* __restrict__ src, const int* __restrict__ dst,
                          int* __restrict__ cursor, int* __restrict__ col, int E) {
  int e = blockIdx.x * blockDim.x + threadIdx.x;
  if (e < E) {
    int d = dst[e];
    int pos = atomicAdd(&cursor[d], 1);
    col[pos] = src[e];
  }
}

// ---------------- fused SpMM + WMMA GEMM layer ----------------
// Block = 512 threads = 16 waves, grid = ceil(N/16).
// Phase 1: wave w pull-aggregates node (blockIdx*16+w) into LDS row w
//          (x * norm_src gathered over CSR neighbors, then * norm_dst).
// Phase 2: first NCOL/16 waves compute the 16xNCOL output tile with
//          chained V_WMMA_F32_16X16X4_F32 (K=128), + bias (+ReLU).
template <int NCOL, bool RELU>
__global__ void k_fused_layer(const float* __restrict__ x,
                              const int* __restrict__ rowptr,
                              const int* __restrict__ col,
                              const float* __restrict__ nsrc,
                              const float* __restrict__ ndst,
                              const float* __restrict__ W,
                              const float* __restrict__ bias,
                              float* __restrict__ out, int n) {
  __shared__ float sA[16 * LDA];

  const int tid  = threadIdx.x;
  const int wid  = tid >> 5;
  const int lane = tid & 31;
  const size_t grow0 = (size_t)blockIdx.x * 16;

  // ---- phase 1: aggregation (one wave per node, float4 per lane) ----
  const int node = (int)grow0 + wid;
  float4 acc; acc.x = acc.y = acc.z = acc.w = 0.0f;
  float nd = 0.0f;
  if (node < n) {
    nd = ndst[node];
    const int beg = rowptr[node];
    const int end = rowptr[node + 1];
    for (int e = beg; e < end; ++e) {
      int s = col[e];
      if (e + 1 < end) {
        // global_prefetch_b8 for the next neighbor's feature row
        __builtin_prefetch(&x[(size_t)col[e + 1] * FEATS + lane * 4], 0, 0);
      }
      float w = nsrc[s];
      float4 v = ((const float4*)(x + (size_t)s * FEATS))[lane];
      acc.x += v.x * w; acc.y += v.y * w; acc.z += v.z * w; acc.w += v.w * w;
    }
  }
  float4 r4; r4.x = acc.x * nd; r4.y = acc.y * nd; r4.z = acc.z * nd; r4.w = acc.w * nd;
  ((float4*)(&sA[wid * LDA]))[lane] = r4;   // row wid, 16B-aligned (LDA*4 = 528)
  __syncthreads();

  // ---- phase 2: WMMA GEMM on the LDS tile ----
  constexpr int CT = NCOL / 16;            // column tiles (8 or 4)
  if (wid < CT) {                          // wave-uniform branch: EXEC all-ones inside
    const int nt   = wid;
    const int l    = lane & 15;
    const int half = lane >> 4;
    const int ncol = nt * 16 + l;

    const float* ar = &sA[l * LDA];
    v8f c = {};
#pragma unroll
    for (int k0 = 0; k0 < FEATS; k0 += 4) {
      const int kk = k0 + 2 * half;        // lanes 16-31 hold K+2 (ISA A layout)
      v2f a, b;
      a.x = ar[kk];
      a.y = ar[kk + 1];
      b.x = W[kk * NCOL + ncol];
      b.y = W[(kk + 1) * NCOL + ncol];
      c = __builtin_amdgcn_wmma_f32_16x16x4_f32(false, a, false, b, (short)0, c,
                                                false, false);
    }

    const float bv = bias[ncol];
    const size_t gm0 = grow0 + 8 * half;   // D layout: VGPR r -> row r / r+8
#pragma unroll
    for (int r = 0; r < 8; ++r) {
      size_t gm = gm0 + r;
      if (gm < (size_t)n) {
        float v = c[r] + bv;
        if (RELU) v = fmaxf(v, 0.0f);
        out[gm * (size_t)NCOL + ncol] = v;
      }
    }
  }
}

// ---------------- orchestration ----------------
extern "C" void kernel_launch(void* const* d_in, const int* in_sizes, int n_in,
                              void* d_out, int out_size, void* d_ws, size_t ws_size,
                              hipStream_t stream) {
  const float* feats = (const float*)d_in[0];
  const int*   src   = (const int*)d_in[1];
  const int*   dst   = (const int*)d_in[2];
  const float* W0    = (const float*)d_in[3];
  const float* b0    = (const float*)d_in[4];
  const float* W1    = (const float*)d_in[5];
  const float* b1    = (const float*)d_in[6];
  const float* W2    = (const float*)d_in[7];
  const float* b2    = (const float*)d_in[8];

  const int N = in_sizes[0] / FEATS;   // 100000
  const int E = in_sizes[1];           // 1600000

  char* ws = (char*)d_ws;
  size_t off = 0;
  auto take = [&](size_t bytes) -> void* {
    void* p = ws + off;
    off += (bytes + 255) & ~(size_t)255;
    return p;
  };

  const int nb = (N + 255) / 256;
  int*   dout_i = (int*)take((size_t)N * 4);
  int*   din_i  = (int*)take((size_t)N * 4);
  float* nsrc   = (float*)take((size_t)N * 4);
  float* ndst   = (float*)take((size_t)N * 4);
  int*   incl   = (int*)take((size_t)N * 4);
  int*   bsum   = (int*)take((size_t)nb * 4);
  int*   rowptr = (int*)take((size_t)(N + 1) * 4);
  int*   cursor = (int*)take((size_t)N * 4);
  int*   col    = (int*)take((size_t)E * 4);
  float* h0     = (float*)take((size_t)N * FEATS * 4);
  float* h1     = (float*)take((size_t)N * FEATS * 4);

  // zero degree histograms (graph-capture-safe memset nodes)
  hipMemsetAsync(dout_i, 0, (size_t)N * 4, stream);
  hipMemsetAsync(din_i, 0, (size_t)N * 4, stream);

  const int eb = (E + 255) / 256;
  k_degrees<<<eb, 256, 0, stream>>>(src, dst, dout_i, din_i, E);
  k_norms<<<nb, 256, 0, stream>>>(dout_i, din_i, nsrc, ndst, N);

  // CSR by destination
  k_scan1<<<nb, 256, 0, stream>>>(din_i, incl, bsum, N);
  k_scan2<<<1, 64, 0, stream>>>(bsum, nb);
  k_scan3<<<nb, 256, 0, stream>>>(din_i, incl, bsum, rowptr, cursor, N);
  k_scatter<<<eb, 256, 0, stream>>>(src, dst, cursor, col, E);

  const int tiles = (N + 15) / 16;         // 6250 fused tiles

  // layer 0: feats -> h0   (aggregate + GEMM(W0) + ReLU)
  k_fused_layer<128, true><<<tiles, 512, 0, stream>>>(
      feats, rowptr, col, nsrc, ndst, W0, b0, h0, N);
  // layer 1: h0 -> h1
  k_fused_layer<128, true><<<tiles, 512, 0, stream>>>(
      h0, rowptr, col, nsrc, ndst, W1, b1, h1, N);
  // layer 2: h1 -> out (64 classes, no relu)
  k_fused_layer<64, false><<<tiles, 512, 0, stream>>>(
      h1, rowptr, col, nsrc, ndst, W2, b2, (float*)d_out, N);
}